// BigFanoutModel_14319420965116
// MI455X (gfx1250) — compile-verified
//
#include <hip/hip_runtime.h>

// ext-vector types matching the WMMA builtin operand shapes for
// v_wmma_f32_16x16x4_f32: A,B = 2 VGPRs f32 per lane; C,D = 8 VGPRs f32.
typedef float v2f __attribute__((ext_vector_type(2)));
typedef float v8f __attribute__((ext_vector_type(8)));

// out[4] = x[1x4] @ (sum_n matrices[n])   (latency-bound, single wave32)
__global__ __launch_bounds__(32) void bigfanout_wmma_kernel(
    const float* __restrict__ x,
    const float* __restrict__ mats,
    float* __restrict__ out,
    int n_chunks /* = n_matrices * 4 float4 chunks */) {
  const int lane = threadIdx.x;  // 0..31, exactly one wave32

  // ---- Phase 1: accumulate Sum_n M[n] ------------------------------------
  // float4 chunk c = floats [4c,4c+4) = matrix (c>>2), row d = (c&3), cols 0..3.
  // Lane stride 32 keeps (c & 3) == (lane & 3) constant, so each lane owns a
  // single row accumulator for d = lane & 3.
  float4 acc = make_float4(0.f, 0.f, 0.f, 0.f);
  const float4* __restrict__ m4 = reinterpret_cast<const float4*>(mats);
  for (int c = lane; c < n_chunks; c += 32) {
    float4 v = m4[c];  // global_load_b128, 16B aligned
    acc.x += v.x; acc.y += v.y; acc.z += v.z; acc.w += v.w;
  }
  // Fold the 8 lanes that share (lane & 3): xor masks 4, 8, 16.
  for (int mask = 4; mask <= 16; mask <<= 1) {
    acc.x += __shfl_xor(acc.x, mask, 32);
    acc.y += __shfl_xor(acc.y, mask, 32);
    acc.z += __shfl_xor(acc.z, mask, 32);
    acc.w += __shfl_xor(acc.w, mask, 32);
  }
  // Lanes 0..3 now each hold the complete row S[d][0..3] for d == lane.
  __shared__ float S[4][4];
  if (lane < 4) {
    S[lane][0] = acc.x; S[lane][1] = acc.y;
    S[lane][2] = acc.z; S[lane][3] = acc.w;
  }
  __syncthreads();  // single wave: just a dscnt wait

  // ---- Phase 2: one v_wmma_f32_16x16x4_f32: D = A(16x4) x B(4x16) --------
  // A layout (ISA 7.12.2, 32-bit A 16x4): lanes 0-15 = M, VGPR0/1 = K0/K1;
  //                                        lanes 16-31 = M, VGPR0/1 = K2/K3.
  // Put x in row M=0 only.
  const int n    = lane & 15;   // M index for A, N (column) index for B/D
  const int half = lane >> 4;   // selects K pair {0,1} vs {2,3}
  const float x0 = x[0], x1 = x[1], x2 = x[2], x3 = x[3];  // uniform loads

  v2f a;
  a.x = (n == 0) ? (half ? x2 : x0) : 0.f;   // K = half?2:0
  a.y = (n == 0) ? (half ? x3 : x1) : 0.f;   // K = half?3:1

  // B layout (4x16, mirrored striping): VGPR0 = row K=(half?2:0) at col N=lane&15,
  //                                     VGPR1 = row K=(half?3:1). Cols >= 4 are 0.
  v2f b;
  b.x = (n < 4) ? S[half ? 2 : 0][n] : 0.f;
  b.y = (n < 4) ? S[half ? 3 : 1][n] : 0.f;

  v8f c = {};  // zero accumulator
  // 8 args: (neg_a, A, neg_b, B, c_mod, C, reuse_a, reuse_b)
  v8f d = __builtin_amdgcn_wmma_f32_16x16x4_f32(
      false, a, false, b, (short)0, c, false, false);

  // D layout: VGPR0, lanes 0-15 = (M=0, N=lane) -> out row is d[0] on lanes 0..3.
  if (lane < 4) out[lane] = d[0];
}

extern "C" void kernel_launch(void* const* d_in, const int* in_sizes, int n_in,
                              void* d_out, int out_size, void* d_ws, size_t ws_size,
                              hipStream_t stream) {
  (void)n_in; (void)out_size; (void)d_ws; (void)ws_size;
  const float* x    = (const float*)d_in[0];   // [1,4] f32
  const float* mats = (const float*)d_in[1];   // [100,4,4] f32
  float* out        = (float*)d_out;           // [4] f32

  const int n_chunks = in_sizes[1] / 4;        // float4 chunks = n_mat * 4
  bigfanout_wmma_kernel<<<1, 32, 0, stream>>>(x, mats, out, n_chunks);
}